// GINEncoder_32633161515327
// MI455X (gfx1250) — compile-verified
//
#include <hip/hip_runtime.h>

#define NN 100000
#define NE 1600000

typedef __attribute__((ext_vector_type(2))) float v2f;
typedef __attribute__((ext_vector_type(8))) float v8f;

__device__ __forceinline__ v8f wmma4(v2f a, v2f b, v8f c) {
  // D = A(16x4) * B(4x16) + C(16x16), fp32 all the way
  return __builtin_amdgcn_wmma_f32_16x16x4_f32(false, a, false, b, (short)0, c,
                                               false, false);
}

__global__ void __launch_bounds__(256) k_copy4(const float4* __restrict__ s,
                                               float4* __restrict__ d, int n) {
  int i = blockIdx.x * 256 + threadIdx.x;
  if (i < n) d[i] = s[i];
}

// Scatter-add: agg[dst] += feat[src], C channels, 4 channels per thread.
template <int C>
__global__ void __launch_bounds__(256) k_scatter(const float* __restrict__ feat,
                                                 const int* __restrict__ srcI,
                                                 const int* __restrict__ dstI,
                                                 float* __restrict__ agg) {
  unsigned tid = blockIdx.x * 256u + threadIdx.x;
  constexpr unsigned G = C / 4;
  unsigned e = tid / G;
  unsigned g = tid % G;
  if (e >= NE) return;
  int s = srcI[e];
  int d = dstI[e];
  float4 v = *(const float4*)(feat + (unsigned)s * C + g * 4u);
  float* p = agg + (unsigned)d * C + g * 4u;
  unsafeAtomicAdd(p + 0, v.x);
  unsafeAtomicAdd(p + 1, v.y);
  unsafeAtomicAdd(p + 2, v.z);
  unsafeAtomicAdd(p + 3, v.w);
}

// Fused GIN MLP: out = (relu?) ( relu(z @ W1 + b1) @ W2 + b2 ),  z: [NN, K1]
// W1: [K1,32], W2: [32,16]. One 16-node tile per wave; 8 waves per block.
// NHEAD==2 runs two weight sets (mu / logstd) sharing the A-tile load.
template <int K1, int NHEAD>
__global__ void __launch_bounds__(256) k_gin_mlp(
    const float* __restrict__ zin,
    const float* __restrict__ W1h0, const float* __restrict__ b1h0,
    const float* __restrict__ W2h0, const float* __restrict__ b2h0,
    float* __restrict__ outh0,
    const float* __restrict__ W1h1, const float* __restrict__ b1h1,
    const float* __restrict__ W2h1, const float* __restrict__ b2h1,
    float* __restrict__ outh1,
    float* __restrict__ outCopy,   // optional duplicate of head-0 output
    int reluOut) {
  __shared__ float lds[8 * 16 * 33];
  const int lane = (int)(threadIdx.x & 31u);
  const int wave = (int)(threadIdx.x >> 5);
  const int tile = blockIdx.x * 8 + wave;
  const int m = lane & 15;   // A-matrix row / D-matrix column index
  const int half = lane >> 4;
  const int n = m;           // B/C/D column owned by this lane
  float* H = lds + wave * (16 * 33);

  // ---- load A tile (16 rows x K1), rows clamped so EXEC stays all-ones ----
  int row = tile * 16 + m;
  if (row >= NN) row = 0;
  const float* zrow = zin + (size_t)row * K1;
  v2f a[K1 / 4];
#pragma unroll
  for (int j = 0; j < K1 / 4; ++j)
    a[j] = *(const v2f*)(zrow + 4 * j + 2 * half);  // K = 4j+2h, 4j+2h+1

#pragma unroll
  for (int head = 0; head < NHEAD; ++head) {
    const float* W1 = head ? W1h1 : W1h0;
    const float* b1 = head ? b1h1 : b1h0;
    const float* W2 = head ? W2h1 : W2h0;
    const float* b2 = head ? b2h1 : b2h0;
    float* outp = head ? outh1 : outh0;

    // ---- GEMM1: [16 x K1] @ [K1 x 32] -> two 16x16 tiles ----
    v8f acc0 = {};
    v8f acc1 = {};
#pragma unroll
    for (int j = 0; j < K1 / 4; ++j) {
      const int k0 = 4 * j + 2 * half;
      v2f blo, bhi;
      blo.x = W1[(k0 + 0) * 32 + n];
      blo.y = W1[(k0 + 1) * 32 + n];
      bhi.x = W1[(k0 + 0) * 32 + 16 + n];
      bhi.y = W1[(k0 + 1) * 32 + 16 + n];
      acc0 = wmma4(a[j], blo, acc0);
      acc1 = wmma4(a[j], bhi, acc1);
    }
    const float bias0 = b1[n];
    const float bias1 = b1[16 + n];
#pragma unroll
    for (int v = 0; v < 8; ++v) {  // C/D layout: VGPR v -> M = v + 8*half
      const int rowm = v + 8 * half;
      H[rowm * 33 + n] = fmaxf(acc0[v] + bias0, 0.0f);
      H[rowm * 33 + 16 + n] = fmaxf(acc1[v] + bias1, 0.0f);
    }
    __syncthreads();  // make hidden tile visible across lanes

    // ---- GEMM2: [16 x 32] @ [32 x 16] ----
    v8f acc = {};
#pragma unroll
    for (int j = 0; j < 8; ++j) {
      const int k0 = 4 * j + 2 * half;
      v2f a2, bv;
      a2.x = H[m * 33 + k0];
      a2.y = H[m * 33 + k0 + 1];
      bv.x = W2[(k0 + 0) * 16 + n];
      bv.y = W2[(k0 + 1) * 16 + n];
      acc = wmma4(a2, bv, acc);
    }
    const float biasO = b2[n];
#pragma unroll
    for (int v = 0; v < 8; ++v) {
      const int rowm = v + 8 * half;
      const int node = tile * 16 + rowm;
      float o = acc[v] + biasO;
      if (reluOut) o = fmaxf(o, 0.0f);
      if (node < NN) {
        outp[(size_t)node * 16 + n] = o;
        if (head == 0 && outCopy) outCopy[(size_t)node * 16 + n] = o;
      }
    }
    __syncthreads();  // protect H before next head overwrites it
  }
}

extern "C" void kernel_launch(void* const* d_in, const int* in_sizes, int n_in,
                              void* d_out, int out_size, void* d_ws,
                              size_t ws_size, hipStream_t stream) {
  const float* x = (const float*)d_in[0];
  const int* ei = (const int*)d_in[1];
  const float* W1_1 = (const float*)d_in[2];
  const float* b1_1 = (const float*)d_in[3];
  const float* W2_1 = (const float*)d_in[4];
  const float* b2_1 = (const float*)d_in[5];
  const float* W1_mu = (const float*)d_in[6];
  const float* b1_mu = (const float*)d_in[7];
  const float* W2_mu = (const float*)d_in[8];
  const float* b2_mu = (const float*)d_in[9];
  const float* W1_ls = (const float*)d_in[10];
  const float* b1_ls = (const float*)d_in[11];
  const float* W2_ls = (const float*)d_in[12];
  const float* b2_ls = (const float*)d_in[13];

  float* out = (float*)d_out;
  char* ws = (char*)d_ws;
  float* agg1 = (float*)(ws);               // [NN,64]  25.6 MB
  float* hbuf = (float*)(ws + 25600000);    // [NN,16]   6.4 MB
  float* agg2 = (float*)(ws + 32000000);    // [NN,16]   6.4 MB
  const int* srcI = ei;
  const int* dstI = ei + NE;

  // agg1 = x (self term), then agg1 += sum_{j->i} x_j
  {
    int n4 = NN * 64 / 4;
    k_copy4<<<(n4 + 255) / 256, 256, 0, stream>>>((const float4*)x,
                                                  (float4*)agg1, n4);
  }
  k_scatter<64><<<(NE * 16) / 256, 256, 0, stream>>>(x, srcI, dstI, agg1);

  const int mlpBlocks = (((NN + 15) / 16) + 7) / 8;  // 782
  // h = relu(MLP1(agg1)); also seed agg2 = h (self term for layer 2/3)
  k_gin_mlp<64, 1><<<mlpBlocks, 256, 0, stream>>>(
      agg1, W1_1, b1_1, W2_1, b2_1, hbuf, W1_1, b1_1, W2_1, b2_1, hbuf, agg2,
      1);
  // agg2 += sum_{j->i} h_j
  k_scatter<16><<<(NE * 4) / 256, 256, 0, stream>>>(hbuf, srcI, dstI, agg2);
  // mu = MLP_mu(agg2), logstd = MLP_ls(agg2)
  k_gin_mlp<16, 2><<<mlpBlocks, 256, 0, stream>>>(
      agg2, W1_mu, b1_mu, W2_mu, b2_mu, out, W1_ls, b1_ls, W2_ls, b2_ls,
      out + (size_t)NN * 16, nullptr, 0);
}